// GMMAttention_46737834115651
// MI455X (gfx1250) — compile-verified
//
#include <hip/hip_runtime.h>
#include <hip/hip_bf16.h>
#include <math.h>

#define NMIX   5
#define BATCH  32
#define LCTX   8192
#define DDIM   256
#define QDIM   512
#define HDIM   256
#define PLEFT  (-50.0f)
#define PRIGHT (50.0f)

typedef __attribute__((ext_vector_type(2))) float v2f;
typedef __attribute__((ext_vector_type(8))) float v8f;

// ---------------------------------------------------------------------------
// Kernel A (1 workgroup, 8 wave32s on one WGP):
//   proj  = tanh(query @ W1^T + b1)   via native fp32 V_WMMA_F32_16X16X4_F32
//   stats = proj @ W2^T + b2          scalar (32x15, trivial)
//   abk   = exp(stats) split into alpha/beta/kappa -> workspace
//
// fp32 WMMA fragment layouts (ISA 7.12.2, 32-bit 16x4 A / 4x16 B):
//   lane = 16*hi + r ; A VGPR v -> A[r][2*hi + v]    (query row, K pair)
//                       B VGPR v -> B[2*hi + v][r]    (W1 row n0+r, K pair)
// Both are one contiguous float2 per lane per K-step of 4 -> no conversions.
// ---------------------------------------------------------------------------
__global__ __launch_bounds__(256) void gmm_stats_kernel(
    const float* __restrict__ query,  // (B, Q)
    const float* __restrict__ W1,     // (H, Q)
    const float* __restrict__ b1,     // (H)
    const float* __restrict__ W2,     // (3*NMIX, H)
    const float* __restrict__ b2,     // (3*NMIX)
    float* __restrict__ abk)          // [3][BATCH][NMIX]
{
    __shared__ float proj_lds[BATCH * HDIM];   // 32 KB

    const int tid  = threadIdx.x;
    const int lane = tid & 31;
    const int wave = tid >> 5;        // 0..7
    const int r    = lane & 15;       // row/col within tile
    const int hi   = lane >> 4;       // half-wave selector

    const int mtile  = wave >> 2;      // 0..1 -> batch rows [16*mtile, 16*mtile+16)
    const int ntbase = (wave & 3) * 4; // 4 N-tiles per wave

    v8f acc[4] = {v8f{}, v8f{}, v8f{}, v8f{}};

    const float* qrow = query + (size_t)(mtile * 16 + r) * QDIM + 2 * hi;
    const float* w1r0 = W1 + (size_t)((ntbase + 0) * 16 + r) * QDIM + 2 * hi;
    const float* w1r1 = W1 + (size_t)((ntbase + 1) * 16 + r) * QDIM + 2 * hi;
    const float* w1r2 = W1 + (size_t)((ntbase + 2) * 16 + r) * QDIM + 2 * hi;
    const float* w1r3 = W1 + (size_t)((ntbase + 3) * 16 + r) * QDIM + 2 * hi;

#pragma unroll 4
    for (int k = 0; k < QDIM; k += 4) {
        // ---- issue all 5 dwordx2 loads first (independent destinations) --
        const float2 aq = *(const float2*)(qrow + k);
        const float2 b0 = *(const float2*)(w1r0 + k);
        const float2 b1f = *(const float2*)(w1r1 + k);
        const float2 b2f = *(const float2*)(w1r2 + k);
        const float2 b3f = *(const float2*)(w1r3 + k);

        const v2f a  = {aq.x, aq.y};
        const v2f f0 = {b0.x, b0.y};
        const v2f f1 = {b1f.x, b1f.y};
        const v2f f2 = {b2f.x, b2f.y};
        const v2f f3 = {b3f.x, b3f.y};

        acc[0] = __builtin_amdgcn_wmma_f32_16x16x4_f32(
            false, a, false, f0, (short)0, acc[0], false, false);
        acc[1] = __builtin_amdgcn_wmma_f32_16x16x4_f32(
            false, a, false, f1, (short)0, acc[1], false, false);
        acc[2] = __builtin_amdgcn_wmma_f32_16x16x4_f32(
            false, a, false, f2, (short)0, acc[2], false, false);
        acc[3] = __builtin_amdgcn_wmma_f32_16x16x4_f32(
            false, a, false, f3, (short)0, acc[3], false, false);
    }

    // ---- bias + tanh, scatter C fragments into LDS (unique slots/lane) ---
#pragma unroll
    for (int t = 0; t < 4; ++t) {
        const int n0 = (ntbase + t) * 16;
#pragma unroll
        for (int v = 0; v < 8; ++v) {
            const int row = mtile * 16 + v + 8 * hi;
            const int col = n0 + r;
            proj_lds[row * HDIM + col] = tanhf(acc[t][v] + b1[col]);
        }
    }
    __syncthreads();

    // ---- stats = proj @ W2^T + b2 ; abk = exp(stats) ---------------------
    for (int idx = tid; idx < BATCH * 3 * NMIX; idx += 256) {
        const int b = idx / (3 * NMIX);
        const int j = idx % (3 * NMIX);
        float s = b2[j];
        const float* pr = proj_lds + b * HDIM;
        const float* wr = W2 + j * HDIM;
        for (int h = 0; h < HDIM; ++h) s += pr[h] * wr[h];
        const int kind = j / NMIX;
        const int m    = j % NMIX;
        abk[kind * BATCH * NMIX + b * NMIX + m] = expf(s);
    }
}

// ---------------------------------------------------------------------------
// Kernel B (32 blocks, one per batch):
//   phase 1: full p_ctx row (mask applied analytically, exact zeros outside)
//   phase 2: expected_ctx over union of prune windows only; ctx loads are
//            skipped wave-uniformly whenever p==0 (p depends only on l).
//            Next chunk is prefetched (global_prefetch_b8) while the current
//            one is reduced.
// ---------------------------------------------------------------------------
__global__ __launch_bounds__(256) void gmm_score_kernel(
    const float* __restrict__ ctx,   // (B, L, D)
    const float* __restrict__ abk,   // [3][BATCH][NMIX]
    float* __restrict__ expected,    // (B, D)
    float* __restrict__ p_out)       // (B, L)
{
    __shared__ float s_abk[3 * NMIX];
    __shared__ float pbuf[512];

    const int b   = blockIdx.x;
    const int tid = threadIdx.x;

    if (tid < 3 * NMIX) {
        const int kind = tid / NMIX, m = tid % NMIX;
        s_abk[tid] = abk[kind * BATCH * NMIX + b * NMIX + m];
    }
    __syncthreads();

    float al[NMIX], be[NMIX], ka[NMIX], lf[NMIX], rf[NMIX];
    int lo = LCTX, hiE = 0;
#pragma unroll
    for (int m = 0; m < NMIX; ++m) {
        al[m] = s_abk[0 * NMIX + m];
        be[m] = s_abk[1 * NMIX + m];
        ka[m] = s_abk[2 * NMIX + m];
        const float c = rintf(ka[m]);          // round-half-to-even, like jnp.round
        lf[m] = c + PLEFT;
        rf[m] = c + PRIGHT + 1.0f;
        const int li = (int)lf[m], ri = (int)rf[m];
        lo  = li < lo  ? li : lo;
        hiE = ri > hiE ? ri : hiE;
    }
    lo  = lo  < 0    ? 0    : lo;
    hiE = hiE > LCTX ? LCTX : hiE;

    // ---- phase 1: p_ctx[b, :] --------------------------------------------
    for (int l = tid; l < LCTX; l += 256) {
        const float fl = (float)l;
        float p = 0.0f;
#pragma unroll
        for (int m = 0; m < NMIX; ++m) {
            if (fl >= lf[m] && fl < rf[m]) {
                const float d = fl - ka[m];
                p += al[m] * expf(-be[m] * d * d);
            }
        }
        p_out[(size_t)b * LCTX + l] = p;
    }

    // ---- phase 2: expected_ctx[b, tid] over the window union -------------
    float accv = 0.0f;
    const int d = tid;                         // 256 threads == DDIM
    for (int base = lo; base < hiE; base += 512) {
        const int n = (hiE - base) < 512 ? (hiE - base) : 512;

        // prefetch the start of the next chunk's ctx rows while we work
        if (base + 512 < hiE)
            __builtin_prefetch(ctx + ((size_t)b * LCTX + base + 512) * DDIM + d, 0, 0);

        __syncthreads();
        for (int j = tid; j < 512; j += 256) {
            float p = 0.0f;
            const int l = base + j;
            if (j < n) {
                const float fl = (float)l;
#pragma unroll
                for (int m = 0; m < NMIX; ++m) {
                    if (fl >= lf[m] && fl < rf[m]) {
                        const float dd = fl - ka[m];
                        p += al[m] * expf(-be[m] * dd * dd);
                    }
                }
            }
            pbuf[j] = p;
        }
        __syncthreads();
        for (int j = 0; j < n; ++j) {
            const float p = pbuf[j];           // uniform across lanes
            if (p != 0.0f)
                accv += p * ctx[((size_t)b * LCTX + base + j) * DDIM + d];
        }
    }
    expected[(size_t)b * DDIM + d] = accv;
}

// ---------------------------------------------------------------------------
extern "C" void kernel_launch(void* const* d_in, const int* in_sizes, int n_in,
                              void* d_out, int out_size, void* d_ws, size_t ws_size,
                              hipStream_t stream) {
    const float* ctx   = (const float*)d_in[0];
    const float* query = (const float*)d_in[1];
    const float* W1    = (const float*)d_in[2];
    const float* b1    = (const float*)d_in[3];
    const float* W2    = (const float*)d_in[4];
    const float* b2    = (const float*)d_in[5];

    float* out       = (float*)d_out;
    float* expected  = out;                       // (B, D) first in return order
    float* p_ctx     = out + BATCH * DDIM;        // (B, L) second
    float* abk       = (float*)d_ws;              // 3*BATCH*NMIX floats

    gmm_stats_kernel<<<1, 256, 0, stream>>>(query, W1, b1, W2, b2, abk);
    gmm_score_kernel<<<BATCH, 256, 0, stream>>>(ctx, abk, expected, p_ctx);
}